// Feature_Engineering_76424648065748
// MI455X (gfx1250) — compile-verified
//
#include <hip/hip_runtime.h>

typedef float v2f __attribute__((ext_vector_type(2)));
typedef float v4f __attribute__((ext_vector_type(4)));

namespace {
constexpr int B_   = 16;
constexpr int CQ   = 7;      // channels per half
constexpr int C2   = 14;     // total channels in x
constexpr int N_   = 32768;  // points (power of two)
constexpr int OUTC = 161;    // 9*7 + 2*49
constexpr int IT   = 4;      // columns per thread (multiple-of-4 alignment)
constexpr int COLS = N_ / IT;            // 8192 = 2^13 threads per batch
}

__device__ __forceinline__ void nt_store4(float* p, v4f v) {
    __builtin_nontemporal_store(v, reinterpret_cast<v4f*>(p));
}

// Each thread produces IT=4 consecutive output columns for one batch.
// ~59 MB read (L2 resident), ~338 MB NT b128 writes -> ~17us HBM floor.
__global__ __launch_bounds__(256, 1)
void feat_kernel(const float* __restrict__ x, float* __restrict__ out) {
    const int gid = blockIdx.x * blockDim.x + threadIdx.x;  // over B_*COLS
    const int b   = gid >> 13;                              // / COLS
    const int k   = (gid & (COLS - 1)) * IT;                // column base, %4==0

    const v2f* __restrict__ xb = reinterpret_cast<const v2f*>(x)
                                 + (size_t)b * C2 * N_;
    float* __restrict__ ob = out + (size_t)b * OUTC * N_ + k;

    const int  kp4   = (k + IT < N_) ? (k + IT) : (N_ - 1); // clamped k+4
    const bool lastT = (k == N_ - IT);                      // owns column N-1

    // Token speculative prefetch to warm L2 ahead of the read stream.
    if (k + 8192 < N_) __builtin_prefetch(xb + k + 8192, 0, 3);

    v2f X1[CQ][IT + 1];   // x1 at k .. k+4 per channel
    v2f X2[CQ][IT + 1];   // x2 at k .. k+4 per channel

#pragma unroll
    for (int c = 0; c < CQ; ++c) {
        const v2f* __restrict__ p1 = xb + (size_t)c * N_;
        const v2f* __restrict__ p2 = xb + (size_t)(c + CQ) * N_;
        const v4f* __restrict__ q1 = reinterpret_cast<const v4f*>(p1);
        const v4f* __restrict__ q2 = reinterpret_cast<const v4f*>(p2);

        // halo: x[k-2], x[k-1] (k==0 lane duplicates x[0] -> dx prepend-zeros)
        v4f m1, m2;
        if (k != 0) {
            m1 = q1[(k >> 1) - 1];
            m2 = q2[(k >> 1) - 1];
        } else {
            const v2f t1 = p1[0], t2 = p2[0];
            m1 = (v4f){t1.x, t1.y, t1.x, t1.y};
            m2 = (v4f){t2.x, t2.y, t2.x, t2.y};
        }
        const v4f a01 = q1[k >> 1];        // x1[k], x1[k+1]
        const v4f a23 = q1[(k >> 1) + 1];  // x1[k+2], x1[k+3]
        const v4f b01 = q2[k >> 1];
        const v4f b23 = q2[(k >> 1) + 1];
        const v2f a4  = p1[kp4];           // x1[k+4] (clamped)
        const v2f b4  = p2[kp4];

        X1[c][0] = (v2f){a01.x, a01.y}; X1[c][1] = (v2f){a01.z, a01.w};
        X1[c][2] = (v2f){a23.x, a23.y}; X1[c][3] = (v2f){a23.z, a23.w};
        X1[c][4] = a4;
        X2[c][0] = (v2f){b01.x, b01.y}; X2[c][1] = (v2f){b01.z, b01.w};
        X2[c][2] = (v2f){b23.x, b23.y}; X2[c][3] = (v2f){b23.z, b23.w};
        X2[c][4] = b4;

        // positions k-2 .. k+3 for both halves
        const v2f P1[6] = { (v2f){m1.x, m1.y}, (v2f){m1.z, m1.w},
                            X1[c][0], X1[c][1], X1[c][2], X1[c][3] };
        const v2f P2[6] = { (v2f){m2.x, m2.y}, (v2f){m2.z, m2.w},
                            X2[c][0], X2[c][1], X2[c][2], X2[c][3] };

        // first differences dx at positions k-1 .. k+3, with norms (once each)
        v2f D1[5], D2[5];
        float N1[5], N2[5];
#pragma unroll
        for (int t = 0; t < 5; ++t) {
            D1[t] = (v2f){P1[t + 1].x - P1[t].x, P1[t + 1].y - P1[t].y};
            D2[t] = (v2f){P2[t + 1].x - P2[t].x, P2[t + 1].y - P2[t].y};
            N1[t] = __builtin_amdgcn_sqrtf(D1[t].x * D1[t].x + D1[t].y * D1[t].y);
            N2[t] = __builtin_amdgcn_sqrtf(D2[t].x * D2[t].x + D2[t].y * D2[t].y);
        }

        float f_n1[IT], f_n2[IT], f_c1[IT], f_c2[IT], f_a1[IT], f_a2[IT];
        float f_di[IT], f_l1[IT], f_l2[IT];
#pragma unroll
        for (int m = 0; m < IT; ++m) {      // column k+m
            const v2f dc1 = D1[m + 1], dp1 = D1[m];
            const v2f dc2 = D2[m + 1], dp2 = D2[m];
            const float n1 = N1[m + 1], n1m = N1[m];
            const float n2 = N2[m + 1], n2m = N2[m];
            f_n1[m] = n1;
            f_n2[m] = n2;
            f_a1[m] = (dc1.x * dp1.x + dc1.y * dp1.y)
                      * __builtin_amdgcn_rcpf(n1 * n1m + 1e-4f);
            f_a2[m] = (dc2.x * dp2.x + dc2.y * dp2.y)
                      * __builtin_amdgcn_rcpf(n2 * n2m + 1e-4f);
            const float dd1x = dc1.x - dp1.x, dd1y = dc1.y - dp1.y;
            const float dd2x = dc2.x - dp2.x, dd2y = dc2.y - dp2.y;
            f_c1[m] = dc1.x * dd1y - dc1.y * dd1x;
            f_c2[m] = dc2.x * dd2y - dc2.y * dd2x;
            f_di[m] = (dc1.x * dc2.x + dc1.y * dc2.y)
                      * __builtin_amdgcn_rcpf(n1 * n2 + 1e-6f);
            const float relx = X1[c][m].x - X2[c][m].x;
            const float rely = X1[c][m].y - X2[c][m].y;
            const float nrel = __builtin_amdgcn_sqrtf(relx * relx + rely * rely);
            f_l1[m] =  (dc1.x * relx + dc1.y * rely)
                      * __builtin_amdgcn_rcpf(n1 * nrel + 1e-6f);
            f_l2[m] = -(dc2.x * relx + dc2.y * rely)
                      * __builtin_amdgcn_rcpf(n2 * nrel + 1e-6f);
        }

        // Channel layout: n1:0 n2:7 cross1:14 cross2:21 adx1:28 adx2:35
        // dirs:42 d:49 dd:98 lead1:147 lead2:154
        nt_store4(ob + (size_t)(  0 + c) * N_, (v4f){f_n1[0], f_n1[1], f_n1[2], f_n1[3]});
        nt_store4(ob + (size_t)(  7 + c) * N_, (v4f){f_n2[0], f_n2[1], f_n2[2], f_n2[3]});
        nt_store4(ob + (size_t)( 14 + c) * N_, (v4f){f_c1[0], f_c1[1], f_c1[2], f_c1[3]});
        nt_store4(ob + (size_t)( 21 + c) * N_, (v4f){f_c2[0], f_c2[1], f_c2[2], f_c2[3]});
        nt_store4(ob + (size_t)( 28 + c) * N_, (v4f){f_a1[0], f_a1[1], f_a1[2], f_a1[3]});
        nt_store4(ob + (size_t)( 35 + c) * N_, (v4f){f_a2[0], f_a2[1], f_a2[2], f_a2[3]});
        nt_store4(ob + (size_t)( 42 + c) * N_, (v4f){f_di[0], f_di[1], f_di[2], f_di[3]});
        nt_store4(ob + (size_t)(147 + c) * N_, (v4f){f_l1[0], f_l1[1], f_l1[2], f_l1[3]});
        nt_store4(ob + (size_t)(154 + c) * N_, (v4f){f_l2[0], f_l2[1], f_l2[2], f_l2[3]});
    }

    // Fold the reference's componentwise +1e-6 into X2 once (saves ~490 adds).
#pragma unroll
    for (int j = 0; j < CQ; ++j) {
#pragma unroll
        for (int p = 0; p <= IT; ++p) {
            X2[j][p].x -= 1e-6f;
            X2[j][p].y -= 1e-6f;
        }
    }

    // 49-pair rolled distance matrix + forward difference.
    // Pair p = i*7 + j pairs x1[(j-i-1) mod 7] with x2[j].
#pragma unroll
    for (int i = 0; i < CQ; ++i) {
#pragma unroll
        for (int j = 0; j < CQ; ++j) {
            const int r  = (j + CQ - 1 - i) % CQ;   // compile-time roll index
            const int ch = i * CQ + j;

            float dv[IT + 1];
#pragma unroll
            for (int p = 0; p <= IT; ++p) {
                const float ex = X1[r][p].x - X2[j][p].x;
                const float ey = X1[r][p].y - X2[j][p].y;
                dv[p] = __builtin_amdgcn_sqrtf(ex * ex + ey * ey);
            }
            const v4f d4  = (v4f){dv[0], dv[1], dv[2], dv[3]};
            const v4f dd4 = (v4f){dv[1] - dv[0], dv[2] - dv[1], dv[3] - dv[2],
                                  lastT ? -dv[3] : dv[4] - dv[3]};
            nt_store4(ob + (size_t)(49 + ch) * N_, d4);
            nt_store4(ob + (size_t)(98 + ch) * N_, dd4);
        }
    }
}

extern "C" void kernel_launch(void* const* d_in, const int* in_sizes, int n_in,
                              void* d_out, int out_size, void* d_ws, size_t ws_size,
                              hipStream_t stream) {
    const float* x = reinterpret_cast<const float*>(d_in[0]);  // (16,14,32768,2) f32
    // d_in[1] (context) is unused by the reference computation.
    float* out = reinterpret_cast<float*>(d_out);              // (16,161,32768) f32

    constexpr int total   = B_ * COLS;        // 131072 threads, 4 columns each
    constexpr int threads = 256;              // 8 wave32 per block
    feat_kernel<<<dim3(total / threads), dim3(threads), 0, stream>>>(x, out);
}